// PINNs_12300786336350
// MI455X (gfx1250) — compile-verified
//
#include <hip/hip_runtime.h>
#include <math.h>

// CDNA5 / gfx1250 PINN kernel: tiny MLP (1-10-10-10-1, tanh) + forward-mode
// 1st and 2nd derivatives, computed with V_WMMA_F32_16X16X4_F32.
//
// Layout per wave (wave32), 16 points per tile:
//   C/D layout (activation streams): VGPR v, lane l ->
//       hidden index i = v + 8*(l>=16), point p = l & 15
//   A operand = W^T (16x4 chunk): lane l holds row i = l&15,
//       VGPR j holds K = 4*chunk + 2*(l>=16) + j
//   B operand = H^T (4x16 chunk): VGPR j, lanes 0-15 hold K = 4c+j,
//       lanes 16-31 hold K = 4c+2+j  (built from C layout via half-swaps)

typedef float v2f __attribute__((ext_vector_type(2)));
typedef float v8f __attribute__((ext_vector_type(8)));

__device__ __forceinline__ float fast_tanh(float x) {
#if __has_builtin(__builtin_amdgcn_tanhf)
    return __builtin_amdgcn_tanhf(x);
#elif __has_builtin(__builtin_amdgcn_tanh_f32)
    return __builtin_amdgcn_tanh_f32(x);
#else
    return tanhf(x);
#endif
}

// Cross half-wave swap: lane l <-> lane l^16.  v_permlanex16_b32 with identity
// lane selects does exactly this in one VALU op (no LDS round trip).
__device__ __forceinline__ float swap_half(float v) {
#if __has_builtin(__builtin_amdgcn_permlanex16)
    int iv = __float_as_int(v);
    int r  = __builtin_amdgcn_permlanex16(iv, iv, 0x76543210, 0xFEDCBA98,
                                          false, false);
    return __int_as_float(r);
#else
    return __shfl_xor(v, 16, 32);
#endif
}

// One stream through one 10x10 layer: returns Z^T = W^T * S^T (C layout, 8 VGPRs)
__device__ __forceinline__ v8f wmma_stream(const float* sc, const float* aw, bool upper) {
    // Shared swap for chunk0-upper (needs sc[2+j] from lower half) and
    // chunk2-lower (needs sc[j] from upper half):
    //   q[j] = upper ? sc[j] : sc[2+j];  s[j] = swap(q[j])
    //   -> lower lanes: s[j] = sc[j](l+16)   (k=8+j for chunk2)
    //   -> upper lanes: s[j] = sc[2+j](l-16) (k=2+j for chunk0)
    float q0 = upper ? sc[0] : sc[2];
    float q1 = upper ? sc[1] : sc[3];
    float s0 = swap_half(q0);
    float s1 = swap_half(q1);
    float s6 = swap_half(sc[6]);   // chunk1-upper: k=6
    float s7 = swap_half(sc[7]);   // chunk1-upper: k=7

    v2f bch0, bch1, bch2;
    // chunk0: lower K=0,1 ; upper K=2,3
    bch0.x = upper ? s0 : sc[0];
    bch0.y = upper ? s1 : sc[1];
    // chunk1: lower K=4,5 ; upper K=6,7
    bch1.x = upper ? s6 : sc[4];
    bch1.y = upper ? s7 : sc[5];
    // chunk2: lower K=8,9 ; upper K=10,11 (zero padding by construction)
    bch2.x = upper ? sc[2] : s0;
    bch2.y = upper ? sc[3] : s1;

    v2f a0 = { aw[0], aw[1] };
    v2f a1 = { aw[2], aw[3] };
    v2f a2 = { aw[4], aw[5] };

    v8f acc = {0.f, 0.f, 0.f, 0.f, 0.f, 0.f, 0.f, 0.f};
    acc = __builtin_amdgcn_wmma_f32_16x16x4_f32(false, a0, false, bch0, (short)0, acc, false, false);
    acc = __builtin_amdgcn_wmma_f32_16x16x4_f32(false, a1, false, bch1, (short)0, acc, false, false);
    acc = __builtin_amdgcn_wmma_f32_16x16x4_f32(false, a2, false, bch2, (short)0, acc, false, false);
    return acc;
}

__global__ void __launch_bounds__(256)
pinn_wmma_kernel(const float* __restrict__ x,
                 const float* __restrict__ W1, const float* __restrict__ b1,
                 const float* __restrict__ W2, const float* __restrict__ b2,
                 const float* __restrict__ W3, const float* __restrict__ b3,
                 const float* __restrict__ W4,
                 float* __restrict__ out, int n)
{
    const int lane   = threadIdx.x & 31;
    const int wave   = (int)((blockIdx.x * blockDim.x + threadIdx.x) >> 5);
    const int nwaves = (int)((gridDim.x * blockDim.x) >> 5);
    const int p      = lane & 15;        // point within tile
    const int hh     = lane >> 4;        // 0 = lower half, 1 = upper half
    const bool upper = (hh != 0);
    const int i      = lane & 15;        // A-matrix row (output hidden unit)

    // ---- load weights once per wave (zero-padded to 16) ----
    float aW2[6], aW3[6];
#pragma unroll
    for (int c = 0; c < 3; ++c) {
#pragma unroll
        for (int j = 0; j < 2; ++j) {
            int k  = 4 * c + 2 * hh + j;           // contraction index
            bool ok = (k < 10) && (i < 10);
            aW2[2 * c + j] = ok ? W2[k * 10 + i] : 0.f;
            aW3[2 * c + j] = ok ? W3[k * 10 + i] : 0.f;
        }
    }
    float W1c[8], b1c[8], b2c[8], b3c[8], W4c[8];
#pragma unroll
    for (int v = 0; v < 8; ++v) {
        int ic = v + 8 * hh;
        bool ok = (ic < 10);
        W1c[v] = ok ? W1[ic] : 0.f;
        b1c[v] = ok ? b1[ic] : 0.f;
        b2c[v] = ok ? b2[ic] : 0.f;
        b3c[v] = ok ? b3[ic] : 0.f;
        W4c[v] = ok ? W4[ic] : 0.f;
    }

    const int ntiles = (n + 15) >> 4;

    for (int t = wave; t < ntiles; t += nwaves) {
        const int base = t << 4;
        const int idx  = base + p;
        float xl = (idx < n) ? x[idx] : 0.f;       // both halves load point p

        // ---- layer 1 (rank-1, elementwise in C layout) ----
        float hc[8], dhc[8], d2hc[8];
#pragma unroll
        for (int v = 0; v < 8; ++v) {
            float w  = W1c[v];
            float z  = fmaf(w, xl, b1c[v]);
            float h  = fast_tanh(z);
            float g  = 1.f - h * h;                // sech^2
            hc[v]   = h;
            dhc[v]  = g * w;                       // dz/dx = W1
            d2hc[v] = -2.f * h * g * w * w;        // z'' = 0
        }

        // ---- layers 2 and 3 (WMMA) ----
#pragma unroll
        for (int layer = 0; layer < 2; ++layer) {
            const float* aw = (layer == 0) ? aW2 : aW3;
            const float* bc = (layer == 0) ? b2c : b3c;
            v8f z   = wmma_stream(hc,   aw, upper);
            v8f dz  = wmma_stream(dhc,  aw, upper);
            v8f d2z = wmma_stream(d2hc, aw, upper);
#pragma unroll
            for (int v = 0; v < 8; ++v) {
                float zz = z[v] + bc[v];
                float h  = fast_tanh(zz);
                float g  = 1.f - h * h;
                float d1 = dz[v];
                hc[v]   = h;
                dhc[v]  = g * d1;
                d2hc[v] = g * d2z[v] - 2.f * h * g * d1 * d1;
            }
        }

        // ---- output layer: dot with W4 over hidden dim (8 local + cross-half) ----
        float u = 0.f, ux = 0.f, uxx = 0.f;
#pragma unroll
        for (int v = 0; v < 8; ++v) {
            u   = fmaf(hc[v],   W4c[v], u);
            ux  = fmaf(dhc[v],  W4c[v], ux);
            uxx = fmaf(d2hc[v], W4c[v], uxx);
        }
        u   += swap_half(u);
        ux  += swap_half(ux);
        uxx += swap_half(uxx);

        if (hh == 0 && idx < n) {
            out[idx * 3 + 0] = u;
            out[idx * 3 + 1] = ux;
            out[idx * 3 + 2] = uxx;
        }
    }
}

extern "C" void kernel_launch(void* const* d_in, const int* in_sizes, int n_in,
                              void* d_out, int out_size, void* d_ws, size_t ws_size,
                              hipStream_t stream) {
    const float* x  = (const float*)d_in[0];
    const float* W1 = (const float*)d_in[1];
    const float* b1 = (const float*)d_in[2];
    const float* W2 = (const float*)d_in[3];
    const float* b2 = (const float*)d_in[4];
    const float* W3 = (const float*)d_in[5];
    const float* b3 = (const float*)d_in[6];
    const float* W4 = (const float*)d_in[7];
    float* out = (float*)d_out;
    int n = in_sizes[0];

    dim3 block(256);
    dim3 grid(1024);   // 8192 waves, grid-stride over n/16 tiles
    pinn_wmma_kernel<<<grid, block, 0, stream>>>(x, W1, b1, W2, b2, W3, b3, W4, out, n);
}